// FullyConnectedSteerableGeometricProductLayer_22454089024084
// MI455X (gfx1250) — compile-verified
//
#include <hip/hip_runtime.h>

// ---------------------------------------------------------------------------
// CDNA5 (gfx1250) wave32 WMMA implementation of the fully-connected steerable
// geometric-product layer (Cl(3,0,0), 8 blades, 20 grade-paths, 64 Cayley
// entries).  All GEMMs run on V_WMMA_F32_16X16X4_F32 in full f32 precision.
//
// Fast path: inputs are repacked once per call into fragment-native layouts
// (k contiguous) so every WMMA operand is a single aligned global_load_b64 —
// no VGPR shuffle movs.  Falls back to the direct-layout kernels if the
// workspace is too small for the repacked copies.
// ---------------------------------------------------------------------------

typedef __attribute__((ext_vector_type(2))) float v2f;
typedef __attribute__((ext_vector_type(8))) float v8f;

#define GA_B    2048
#define GA_NIN  512
#define GA_NOUT 512

__device__ __forceinline__ v2f mk2(float a, float b) { v2f r; r.x = a; r.y = b; return r; }

__device__ __forceinline__ v8f wmma_f32(v2f a, v2f b, v8f c) {
  // D = A(16x4) * B(4x16) + C(16x16), f32 in / f32 out
  return __builtin_amdgcn_wmma_f32_16x16x4_f32(false, a, false, b, (short)0, c, false, false);
}

// blade -> grade (blades: 1,e1,e2,e3,e12,e13,e23,e123)
static constexpr int GRADE_[8] = {0, 1, 1, 1, 2, 2, 2, 3};

// 64 nonzero Cayley entries, enumerated in the reference's path order.
// entry e: out[:, :, EBJ] += (x[:,:,EBI] * xr[:,:,EBK] * ECV) @ weight[:,:,EPX]^T
static constexpr int EBI[64] = {
  0,0,0,0,0,0,0,0,
  1,2,3, 1,2,3,
  1,1,2,2,3,3,
  1,1,2,2,3,3,
  1,2,3, 1,2,3,
  4,5,6,
  4,4,5,5,6,6,
  4,5,6, 4,5,6,
  4,4,5,5,6,6,
  4,5,6,
  7, 7,7,7, 7,7,7, 7};
static constexpr int EBJ[64] = {
  0,1,2,3,4,5,6,7,
  0,0,0, 1,2,3,
  2,3,1,3,1,2,
  4,5,4,6,5,6,
  6,5,4, 7,7,7,
  0,0,0,
  1,2,1,3,2,3,
  3,2,1, 4,5,6,
  5,6,4,6,4,5,
  7,7,7,
  0, 1,2,3, 4,5,6, 7};
static constexpr int EBK[64] = {
  0,1,2,3,4,5,6,7,
  1,2,3, 0,0,0,
  4,5,4,6,5,6,
  2,3,1,3,1,2,
  7,7,7, 6,5,4,
  4,5,6,
  2,1,3,1,3,2,
  7,7,7, 0,0,0,
  6,5,6,4,5,4,
  3,2,1,
  7, 6,5,4, 3,2,1, 0};
static constexpr int ECV[64] = {
  1,1,1,1,1,1,1,1,
  1,1,1, 1,1,1,
  1,1,-1,1,-1,-1,
  1,1,-1,1,-1,-1,
  1,-1,1, 1,-1,1,
  -1,-1,-1,
  1,-1,1,-1,1,-1,
  -1,1,-1, 1,1,1,
  1,-1,-1,1,1,-1,
  1,-1,1,
  -1, -1,1,-1, 1,-1,1, 1};
static constexpr int EPX[64] = {
  0,1,1,1,2,2,2,3,
  4,4,4, 5,5,5,
  6,6,6,6,6,6,
  7,7,7,7,7,7,
  8,8,8, 9,9,9,
  10,10,10,
  11,11,11,11,11,11,
  12,12,12, 13,13,13,
  14,14,14,14,14,14,
  15,15,15,
  16, 17,17,17, 18,18,18, 19};

// ===========================================================================
// Repack kernels (fast path): put K (the n index) contiguous per channel.
// ===========================================================================

// x (b, n, i) -> x_t (b, i, n): coalesced reads and writes.
__global__ __launch_bounds__(256) void ga_repack_x(
    const float* __restrict__ in, float* __restrict__ outp) {
  const int tid = blockIdx.x * 256 + threadIdx.x;  // = b*512 + n
  const int b = tid >> 9;
  const int n = tid & 511;
  const float4 p0 = *(const float4*)(in + (size_t)tid * 8);
  const float4 p1 = *(const float4*)(in + (size_t)tid * 8 + 4);
  float* dst = outp + (size_t)b * (8 * 512) + n;
  dst[0 * 512] = p0.x; dst[1 * 512] = p0.y; dst[2 * 512] = p0.z; dst[3 * 512] = p0.w;
  dst[4 * 512] = p1.x; dst[5 * 512] = p1.y; dst[6 * 512] = p1.z; dst[7 * 512] = p1.w;
}

// w (m, n, C) -> w_t (m, C, n)
template <int C>
__global__ __launch_bounds__(256) void ga_repack_w(
    const float* __restrict__ in, float* __restrict__ outp) {
  const int tid = blockIdx.x * 256 + threadIdx.x;  // = m*512 + n
  const int m = tid >> 9;
  const int n = tid & 511;
  const float* src = in + (size_t)tid * C;
  float* dst = outp + (size_t)m * (C * 512) + n;
#pragma unroll
  for (int c = 0; c < C; ++c) dst[(size_t)c * 512] = src[c];
}

// ===========================================================================
// Fast kernels: every WMMA operand is a single aligned b64 load.
// ===========================================================================

__global__ __launch_bounds__(32) void ga_xr_norm_fast(
    const float* __restrict__ xt, const float* __restrict__ wrt,
    const float* __restrict__ norm_a, float* __restrict__ xrt) {
  const int b0 = blockIdx.x << 4;
  const int m0 = blockIdx.y << 4;
  const int lane = threadIdx.x & 31;
  const int lo = lane & 15;
  const int hi = lane >> 4;

  const v8f vzero = {};
  v8f acc[8];
#pragma unroll
  for (int i = 0; i < 8; ++i) acc[i] = vzero;

  const float* xrow = xt + (size_t)(b0 + lo) * (8 * 512);   // + i*512 + kk
  const float* wrow = wrt + (size_t)(m0 + lo) * (4 * 512);  // + g*512 + kk

#pragma unroll 4
  for (int k = 0; k < GA_NIN; k += 4) {
    const int kk = k + (hi << 1);
    v2f a[8], g[4];
#pragma unroll
    for (int i = 0; i < 8; ++i) a[i] = *(const v2f*)(xrow + i * 512 + kk);
#pragma unroll
    for (int j = 0; j < 4; ++j) g[j] = *(const v2f*)(wrow + j * 512 + kk);
#pragma unroll
    for (int i = 0; i < 8; ++i)
      acc[i] = wmma_f32(a[i], g[GRADE_[i]], acc[i]);
  }

  // Gating: all 8 blade values of one (b,m) element live in the same lane/slot.
  const float4 na = *(const float4*)(norm_a + (size_t)(m0 + lo) * 4);
  const float s0 = 1.0f / (1.0f + __expf(-na.x));
  const float s1 = 1.0f / (1.0f + __expf(-na.y));
  const float s2 = 1.0f / (1.0f + __expf(-na.z));
  const float s3 = 1.0f / (1.0f + __expf(-na.w));

#pragma unroll
  for (int v = 0; v < 8; ++v) {
    const float e0 = acc[0][v], e1 = acc[1][v], e2 = acc[2][v], e3 = acc[3][v];
    const float e4 = acc[4][v], e5 = acc[5][v], e6 = acc[6][v], e7 = acc[7][v];
    const float n0 = __builtin_sqrtf(e0 * e0);
    const float n1 = __builtin_sqrtf(e1 * e1 + e2 * e2 + e3 * e3);
    const float n2 = __builtin_sqrtf(e4 * e4 + e5 * e5 + e6 * e6);
    const float n3 = __builtin_sqrtf(e7 * e7);
    const float i0 = 1.0f / (s0 * (n0 - 1.0f) + 1.0f + 1e-6f);
    const float i1 = 1.0f / (s1 * (n1 - 1.0f) + 1.0f + 1e-6f);
    const float i2 = 1.0f / (s2 * (n2 - 1.0f) + 1.0f + 1e-6f);
    const float i3 = 1.0f / (s3 * (n3 - 1.0f) + 1.0f + 1e-6f);
    // store in transposed (b, blade, n) layout for kernel 2's A-fragments
    float* dst = xrt + (size_t)(b0 + v + (hi << 3)) * (8 * 512) + (m0 + lo);
    dst[0 * 512] = e0 * i0;
    dst[1 * 512] = e1 * i1;
    dst[2 * 512] = e2 * i1;
    dst[3 * 512] = e3 * i1;
    dst[4 * 512] = e4 * i2;
    dst[5 * 512] = e5 * i2;
    dst[6 * 512] = e6 * i2;
    dst[7 * 512] = e7 * i3;
  }
}

__global__ __launch_bounds__(32) void ga_out_fast(
    const float* __restrict__ xt, const float* __restrict__ xrt,
    const float* __restrict__ wt, const float* __restrict__ wlt,
    const float* __restrict__ b_left, float* __restrict__ out) {
  const int b0 = blockIdx.x << 4;
  const int m0 = blockIdx.y << 4;
  const int lane = threadIdx.x & 31;
  const int lo = lane & 15;
  const int hi = lane >> 4;

  const v8f vzero = {};
  v8f acc[8];
#pragma unroll
  for (int i = 0; i < 8; ++i) acc[i] = vzero;

  const float* xrow = xt + (size_t)(b0 + lo) * (8 * 512);
  const float* rrow = xrt + (size_t)(b0 + lo) * (8 * 512);
  const float* lrow = wlt + (size_t)(m0 + lo) * (4 * 512);
  const float* wrow = wt + (size_t)(m0 + lo) * (20 * 512);

#pragma unroll 1
  for (int k = 0; k < GA_NIN; k += 4) {
    const int kk = k + (hi << 1);
    v2f a[8], r[8], bl[4], bw[20];
#pragma unroll
    for (int i = 0; i < 8; ++i) a[i] = *(const v2f*)(xrow + i * 512 + kk);
#pragma unroll
    for (int i = 0; i < 8; ++i) r[i] = *(const v2f*)(rrow + i * 512 + kk);
#pragma unroll
    for (int j = 0; j < 4; ++j) bl[j] = *(const v2f*)(lrow + j * 512 + kk);
#pragma unroll
    for (int p = 0; p < 20; ++p) bw[p] = *(const v2f*)(wrow + p * 512 + kk);

    // left linear path (grades share weights)
#pragma unroll
    for (int i = 0; i < 8; ++i)
      acc[i] = wmma_f32(a[i], bl[GRADE_[i]], acc[i]);

    // bilinear geometric-product path: 64 Cayley entries
#pragma unroll
    for (int e = 0; e < 64; ++e) {
      v2f p = a[EBI[e]] * r[EBK[e]];  // v_pk_mul_f32
      if (ECV[e] < 0) p = -p;         // sign folded at compile time
      acc[EBJ[e]] = wmma_f32(p, bw[EPX[e]], acc[EBJ[e]]);
    }
  }

  const float bias = b_left[m0 + lo];
  const float rs2 = 0.70710678118654752440f;
#pragma unroll
  for (int v = 0; v < 8; ++v) {
    const float o0 = (acc[0][v] + bias) * rs2;
    const float o1 = acc[1][v] * rs2, o2 = acc[2][v] * rs2, o3 = acc[3][v] * rs2;
    const float o4 = acc[4][v] * rs2, o5 = acc[5][v] * rs2;
    const float o6 = acc[6][v] * rs2, o7 = acc[7][v] * rs2;
    float* dst = out + ((size_t)(b0 + v + (hi << 3)) * GA_NOUT + (m0 + lo)) * 8;
    *(float4*)(dst)     = make_float4(o0, o1, o2, o3);
    *(float4*)(dst + 4) = make_float4(o4, o5, o6, o7);
  }
}

// ===========================================================================
// Fallback kernels (direct layout, need only 32 MB of workspace).
// ===========================================================================

__global__ __launch_bounds__(32) void ga_xr_norm_kernel(
    const float* __restrict__ x, const float* __restrict__ w_right,
    const float* __restrict__ norm_a, float* __restrict__ xr) {
  const int b0 = blockIdx.x << 4;
  const int m0 = blockIdx.y << 4;
  const int lane = threadIdx.x & 31;
  const int lo = lane & 15;
  const int hi = lane >> 4;

  const v8f vzero = {};
  v8f acc[8];
#pragma unroll
  for (int i = 0; i < 8; ++i) acc[i] = vzero;

  const float* xrow = x + (size_t)(b0 + lo) * (GA_NIN * 8);
  const float* wrow = w_right + (size_t)(m0 + lo) * (GA_NIN * 4);

#pragma unroll 2
  for (int k = 0; k < GA_NIN; k += 4) {
    const int kk = k + (hi << 1);
    const float4 xa0 = *(const float4*)(xrow + kk * 8);
    const float4 xa1 = *(const float4*)(xrow + kk * 8 + 4);
    const float4 xb0 = *(const float4*)(xrow + kk * 8 + 8);
    const float4 xb1 = *(const float4*)(xrow + kk * 8 + 12);
    const float4 w0  = *(const float4*)(wrow + kk * 4);
    const float4 w1  = *(const float4*)(wrow + kk * 4 + 4);

    const v2f a0 = mk2(xa0.x, xb0.x), a1 = mk2(xa0.y, xb0.y);
    const v2f a2 = mk2(xa0.z, xb0.z), a3 = mk2(xa0.w, xb0.w);
    const v2f a4 = mk2(xa1.x, xb1.x), a5 = mk2(xa1.y, xb1.y);
    const v2f a6 = mk2(xa1.z, xb1.z), a7 = mk2(xa1.w, xb1.w);
    const v2f g0 = mk2(w0.x, w1.x), g1 = mk2(w0.y, w1.y);
    const v2f g2 = mk2(w0.z, w1.z), g3 = mk2(w0.w, w1.w);

    acc[0] = wmma_f32(a0, g0, acc[0]);
    acc[1] = wmma_f32(a1, g1, acc[1]);
    acc[2] = wmma_f32(a2, g1, acc[2]);
    acc[3] = wmma_f32(a3, g1, acc[3]);
    acc[4] = wmma_f32(a4, g2, acc[4]);
    acc[5] = wmma_f32(a5, g2, acc[5]);
    acc[6] = wmma_f32(a6, g2, acc[6]);
    acc[7] = wmma_f32(a7, g3, acc[7]);
  }

  const float4 na = *(const float4*)(norm_a + (size_t)(m0 + lo) * 4);
  const float s0 = 1.0f / (1.0f + __expf(-na.x));
  const float s1 = 1.0f / (1.0f + __expf(-na.y));
  const float s2 = 1.0f / (1.0f + __expf(-na.z));
  const float s3 = 1.0f / (1.0f + __expf(-na.w));

#pragma unroll
  for (int v = 0; v < 8; ++v) {
    const float e0 = acc[0][v], e1 = acc[1][v], e2 = acc[2][v], e3 = acc[3][v];
    const float e4 = acc[4][v], e5 = acc[5][v], e6 = acc[6][v], e7 = acc[7][v];
    const float n0 = __builtin_sqrtf(e0 * e0);
    const float n1 = __builtin_sqrtf(e1 * e1 + e2 * e2 + e3 * e3);
    const float n2 = __builtin_sqrtf(e4 * e4 + e5 * e5 + e6 * e6);
    const float n3 = __builtin_sqrtf(e7 * e7);
    const float i0 = 1.0f / (s0 * (n0 - 1.0f) + 1.0f + 1e-6f);
    const float i1 = 1.0f / (s1 * (n1 - 1.0f) + 1.0f + 1e-6f);
    const float i2 = 1.0f / (s2 * (n2 - 1.0f) + 1.0f + 1e-6f);
    const float i3 = 1.0f / (s3 * (n3 - 1.0f) + 1.0f + 1e-6f);
    float* dst = xr + ((size_t)(b0 + v + (hi << 3)) * GA_NIN + (m0 + lo)) * 8;
    *(float4*)(dst)     = make_float4(e0 * i0, e1 * i1, e2 * i1, e3 * i1);
    *(float4*)(dst + 4) = make_float4(e4 * i2, e5 * i2, e6 * i2, e7 * i3);
  }
}

__global__ __launch_bounds__(32) void ga_out_kernel(
    const float* __restrict__ x, const float* __restrict__ xr,
    const float* __restrict__ weight, const float* __restrict__ w_left,
    const float* __restrict__ b_left, float* __restrict__ out) {
  const int b0 = blockIdx.x << 4;
  const int m0 = blockIdx.y << 4;
  const int lane = threadIdx.x & 31;
  const int lo = lane & 15;
  const int hi = lane >> 4;

  const v8f vzero = {};
  v8f acc[8];
#pragma unroll
  for (int i = 0; i < 8; ++i) acc[i] = vzero;

  const float* xrow = x  + (size_t)(b0 + lo) * (GA_NIN * 8);
  const float* rrow = xr + (size_t)(b0 + lo) * (GA_NIN * 8);
  const float* lrow = w_left + (size_t)(m0 + lo) * (GA_NIN * 4);
  const float* wrow = weight + (size_t)(m0 + lo) * (GA_NIN * 20);

#pragma unroll 1
  for (int k = 0; k < GA_NIN; k += 4) {
    const int kk = k + (hi << 1);

    const float4 xa0 = *(const float4*)(xrow + kk * 8);
    const float4 xa1 = *(const float4*)(xrow + kk * 8 + 4);
    const float4 xb0 = *(const float4*)(xrow + kk * 8 + 8);
    const float4 xb1 = *(const float4*)(xrow + kk * 8 + 12);
    const float4 ra0 = *(const float4*)(rrow + kk * 8);
    const float4 ra1 = *(const float4*)(rrow + kk * 8 + 4);
    const float4 rb0 = *(const float4*)(rrow + kk * 8 + 8);
    const float4 rb1 = *(const float4*)(rrow + kk * 8 + 12);
    const float4 l0  = *(const float4*)(lrow + kk * 4);
    const float4 l1  = *(const float4*)(lrow + kk * 4 + 4);
    const float4 q0  = *(const float4*)(wrow + kk * 20);
    const float4 q1  = *(const float4*)(wrow + kk * 20 + 4);
    const float4 q2  = *(const float4*)(wrow + kk * 20 + 8);
    const float4 q3  = *(const float4*)(wrow + kk * 20 + 12);
    const float4 q4  = *(const float4*)(wrow + kk * 20 + 16);
    const float4 t0  = *(const float4*)(wrow + kk * 20 + 20);
    const float4 t1  = *(const float4*)(wrow + kk * 20 + 24);
    const float4 t2  = *(const float4*)(wrow + kk * 20 + 28);
    const float4 t3  = *(const float4*)(wrow + kk * 20 + 32);
    const float4 t4  = *(const float4*)(wrow + kk * 20 + 36);

    v2f a[8], r[8], bl[4], bw[20];
    a[0] = mk2(xa0.x, xb0.x); a[1] = mk2(xa0.y, xb0.y);
    a[2] = mk2(xa0.z, xb0.z); a[3] = mk2(xa0.w, xb0.w);
    a[4] = mk2(xa1.x, xb1.x); a[5] = mk2(xa1.y, xb1.y);
    a[6] = mk2(xa1.z, xb1.z); a[7] = mk2(xa1.w, xb1.w);
    r[0] = mk2(ra0.x, rb0.x); r[1] = mk2(ra0.y, rb0.y);
    r[2] = mk2(ra0.z, rb0.z); r[3] = mk2(ra0.w, rb0.w);
    r[4] = mk2(ra1.x, rb1.x); r[5] = mk2(ra1.y, rb1.y);
    r[6] = mk2(ra1.z, rb1.z); r[7] = mk2(ra1.w, rb1.w);
    bl[0] = mk2(l0.x, l1.x); bl[1] = mk2(l0.y, l1.y);
    bl[2] = mk2(l0.z, l1.z); bl[3] = mk2(l0.w, l1.w);
    bw[0]  = mk2(q0.x, t0.x); bw[1]  = mk2(q0.y, t0.y);
    bw[2]  = mk2(q0.z, t0.z); bw[3]  = mk2(q0.w, t0.w);
    bw[4]  = mk2(q1.x, t1.x); bw[5]  = mk2(q1.y, t1.y);
    bw[6]  = mk2(q1.z, t1.z); bw[7]  = mk2(q1.w, t1.w);
    bw[8]  = mk2(q2.x, t2.x); bw[9]  = mk2(q2.y, t2.y);
    bw[10] = mk2(q2.z, t2.z); bw[11] = mk2(q2.w, t2.w);
    bw[12] = mk2(q3.x, t3.x); bw[13] = mk2(q3.y, t3.y);
    bw[14] = mk2(q3.z, t3.z); bw[15] = mk2(q3.w, t3.w);
    bw[16] = mk2(q4.x, t4.x); bw[17] = mk2(q4.y, t4.y);
    bw[18] = mk2(q4.z, t4.z); bw[19] = mk2(q4.w, t4.w);

#pragma unroll
    for (int i = 0; i < 8; ++i)
      acc[i] = wmma_f32(a[i], bl[GRADE_[i]], acc[i]);

#pragma unroll
    for (int e = 0; e < 64; ++e) {
      v2f p = a[EBI[e]] * r[EBK[e]];
      if (ECV[e] < 0) p = -p;
      acc[EBJ[e]] = wmma_f32(p, bw[EPX[e]], acc[EBJ[e]]);
    }
  }

  const float bias = b_left[m0 + lo];
  const float rs2 = 0.70710678118654752440f;
#pragma unroll
  for (int v = 0; v < 8; ++v) {
    const float o0 = (acc[0][v] + bias) * rs2;
    const float o1 = acc[1][v] * rs2, o2 = acc[2][v] * rs2, o3 = acc[3][v] * rs2;
    const float o4 = acc[4][v] * rs2, o5 = acc[5][v] * rs2;
    const float o6 = acc[6][v] * rs2, o7 = acc[7][v] * rs2;
    float* dst = out + ((size_t)(b0 + v + (hi << 3)) * GA_NOUT + (m0 + lo)) * 8;
    *(float4*)(dst)     = make_float4(o0, o1, o2, o3);
    *(float4*)(dst + 4) = make_float4(o4, o5, o6, o7);
  }
}

// ===========================================================================
// Launcher
// ===========================================================================

extern "C" void kernel_launch(void* const* d_in, const int* in_sizes, int n_in,
                              void* d_out, int out_size, void* d_ws, size_t ws_size,
                              hipStream_t stream) {
  (void)in_sizes; (void)n_in; (void)out_size;
  const float* x       = (const float*)d_in[0];  // (2048, 512, 8)
  const float* weight  = (const float*)d_in[1];  // (512, 512, 20)
  const float* w_right = (const float*)d_in[2];  // (512, 512, 4)
  const float* w_left  = (const float*)d_in[3];  // (512, 512, 4)
  const float* b_left  = (const float*)d_in[4];  // (512,)
  const float* norm_a  = (const float*)d_in[5];  // (1, 512, 4)
  float* out = (float*)d_out;                    // (2048, 512, 8)
  float* ws = (float*)d_ws;

  const size_t XT_N  = (size_t)GA_B * 8 * 512;    //  8,388,608 floats
  const size_t W4_N  = (size_t)512 * 4 * 512;     //  1,048,576 floats
  const size_t W20_N = (size_t)512 * 20 * 512;    //  5,242,880 floats
  const size_t FAST_FLOATS = XT_N * 2 + W4_N * 2 + W20_N;  // ~92 MB

  dim3 block(32, 1, 1);
  dim3 grid1(GA_B / 16, GA_NIN / 16, 1);
  dim3 grid2(GA_B / 16, GA_NOUT / 16, 1);

  if (ws_size >= FAST_FLOATS * sizeof(float)) {
    float* xt  = ws;
    float* xrt = xt + XT_N;
    float* wrt = xrt + XT_N;
    float* wlt = wrt + W4_N;
    float* wtt = wlt + W4_N;

    ga_repack_x<<<dim3((GA_B * 512) / 256), dim3(256), 0, stream>>>(x, xt);
    ga_repack_w<4><<<dim3((512 * 512) / 256), dim3(256), 0, stream>>>(w_right, wrt);
    ga_repack_w<4><<<dim3((512 * 512) / 256), dim3(256), 0, stream>>>(w_left, wlt);
    ga_repack_w<20><<<dim3((512 * 512) / 256), dim3(256), 0, stream>>>(weight, wtt);

    ga_xr_norm_fast<<<grid1, block, 0, stream>>>(xt, wrt, norm_a, xrt);
    ga_out_fast<<<grid2, block, 0, stream>>>(xt, xrt, wtt, wlt, b_left, out);
  } else {
    float* xr = ws;  // 32 MB scratch: normalized xr (direct layout)
    ga_xr_norm_kernel<<<grid1, block, 0, stream>>>(x, w_right, norm_a, xr);
    ga_out_kernel<<<grid2, block, 0, stream>>>(x, xr, weight, w_left, b_left, out);
  }
}